// YoloLoss_42004780155205
// MI455X (gfx1250) — compile-verified
//
#include <hip/hip_runtime.h>
#include <cstdint>

// ---------------------------------------------------------------------------
// YOLO loss on MI455X (gfx1250).
// Memory-bound: 252 MB read / 23.3 TB/s  => ~10.8 us floor. Strategy:
//  * async global->LDS staging (CDNA5 GLOBAL_LOAD_ASYNC_TO_LDS_B128, ASYNCcnt
//    double-buffer pipeline) to turn the 120-byte cell records into
//    coalesced streaming loads; pred/target fused into one b128 chunk stream
//    via per-lane address select (async loads take per-lane addresses),
//  * per-cell loss in VALU from LDS (conflict-free: stride 30 dwords,
//    gcd(30,64)=2 -> 32 distinct banks per wave32),
//  * deterministic 2-stage reduction; the 128->1 block reduction uses
//    v_wmma_f32_16x16x4_f32 with an all-ones B matrix (D = A x 1 sums K).
// ---------------------------------------------------------------------------

#define S_DIM 1024
#define BOXES 2
#define NPRED 5
#define CH 30                         // floats per cell
#define CELLS (S_DIM * S_DIM)         // 1,048,576
#define TILE_CELLS 128
#define TILE_FLOATS (TILE_CELLS * CH) // 3840 floats = 15360 B per array
#define TILE_CH16 (TILE_FLOATS / 4)   // 960 16-byte chunks per array
#define BLOCK 128
#define GRID 2048
#define NTILES (CELLS / TILE_CELLS)       // 8192
#define TILES_PER_BLOCK (NTILES / GRID)   // 4

typedef float v2f __attribute__((ext_vector_type(2)));
typedef float v8f __attribute__((ext_vector_type(8)));
typedef int v2i __attribute__((ext_vector_type(2)));
typedef int v4i __attribute__((ext_vector_type(4)));

// Address-space-qualified vector pointer types matching the builtins'
// signatures: global arg is AS1 ("__device__"), LDS arg is AS3 ("__shared__").
typedef v2i __attribute__((address_space(1))) as1_v2i;
typedef v2i __attribute__((address_space(3))) as3_v2i;
typedef v4i __attribute__((address_space(1))) as1_v4i;
typedef v4i __attribute__((address_space(3))) as3_v4i;

#if defined(__has_builtin)
#if __has_builtin(__builtin_amdgcn_global_load_async_to_lds_b128)
#define HAVE_ASYNC_B128 1
#endif
#if __has_builtin(__builtin_amdgcn_global_load_async_to_lds_b64)
#define HAVE_ASYNC_B64 1
#endif
#if __has_builtin(__builtin_amdgcn_s_wait_asynccnt)
#define HAVE_WAIT_BUILTIN 1
#endif
#endif

// async instructions per wave per tile (must match issue_tile)
#if defined(HAVE_ASYNC_B128)
#define ASYNC_PER_TILE 15 // 1920 fused b128 chunks / 128 threads
#else
#define ASYNC_PER_TILE 30 // 15 iters x 2 arrays, b64
#endif

#if defined(HAVE_WAIT_BUILTIN)
#define WAIT_ASYNCCNT(n) __builtin_amdgcn_s_wait_asynccnt(n)
#else
#define WAIT_ASYNCCNT(n) asm volatile("s_wait_asynccnt %0" ::"n"(n) : "memory")
#endif

__device__ __forceinline__ void async_copy_b128(const float* g, float* l) {
#if defined(HAVE_ASYNC_B128)
  as1_v4i* gp = (as1_v4i*)(unsigned long long)(uintptr_t)g;
  as3_v4i* lp = (as3_v4i*)(unsigned)(uintptr_t)l;
  __builtin_amdgcn_global_load_async_to_lds_b128(gp, lp, 0, 0);
#else
  uint32_t laddr = (uint32_t)(uintptr_t)l; // LDS byte offset
  asm volatile("global_load_async_to_lds_b128 %0, %1, off"
               :: "v"(laddr), "v"(g)
               : "memory");
#endif
}

__device__ __forceinline__ void async_copy_b64(const float* g, float* l) {
#if defined(HAVE_ASYNC_B64)
  as1_v2i* gp = (as1_v2i*)(unsigned long long)(uintptr_t)g;
  as3_v2i* lp = (as3_v2i*)(unsigned)(uintptr_t)l;
  __builtin_amdgcn_global_load_async_to_lds_b64(gp, lp, 0, 0);
#else
  uint32_t laddr = (uint32_t)(uintptr_t)l; // LDS byte offset
  asm volatile("global_load_async_to_lds_b64 %0, %1, off"
               :: "v"(laddr), "v"(g)
               : "memory");
#endif
}

// Issue all async loads for one tile: uniform 15 iterations per thread.
// b128 path: pred (chunks 0..959) and target (960..1919) fused into one
// per-lane-selected address stream -> 15 b128 async instrs per wave per tile.
__device__ __forceinline__ void issue_tile(const float* __restrict__ pred,
                                           const float* __restrict__ tgt,
                                           int tile, float* lp, float* lt,
                                           int tid) {
  const float* gp = pred + (size_t)tile * TILE_FLOATS;
  const float* gt = tgt + (size_t)tile * TILE_FLOATS;
#if defined(HAVE_ASYNC_B128)
#pragma unroll
  for (int ch = 0; ch < 2 * TILE_CH16 / BLOCK; ++ch) { // 15
    int i = ch * BLOCK + tid;                          // [0, 1920)
    bool isp = i < TILE_CH16;
    int j = isp ? i : (i - TILE_CH16);
    const float* g = (isp ? gp : gt) + j * 4; // 16B-aligned
    float* l = (isp ? lp : lt) + j * 4;       // 16B-aligned
    async_copy_b128(g, l);
  }
#else
#pragma unroll
  for (int ch = 0; ch < TILE_FLOATS / 2 / BLOCK; ++ch) { // 15
    int i = (ch * BLOCK + tid) * 2; // float index, 8B aligned
    async_copy_b64(gp + i, lp + i);
    async_copy_b64(gt + i, lt + i);
  }
#endif
}

// Per-cell loss with weights folded in.
// conf + noobj terms combine: a*ct + a*(1-ct) == a = (ct-cp)^2.
__device__ __forceinline__ float cell_loss(const float* sp, const float* st) {
  float loss = 0.0f;
#pragma unroll
  for (int b = 0; b < BOXES; ++b) {
    const float* bp = sp + b * NPRED;
    const float* bt = st + b * NPRED;
    float cp = bp[0], xp = bp[1], yp = bp[2];
    float wp = sqrtf(fabsf(bp[3]));
    float hp = sqrtf(fabsf(bp[4]));
    float ct = bt[0], xt = bt[1], yt = bt[2];
    float wt = sqrtf(bt[3]);
    float ht = sqrtf(bt[4]);
    float dx = xt - xp, dy = yt - yp;
    float dw = wt - wp, dh = ht - hp;
    float dc = ct - cp;
    loss += 5.0f * ((dx * dx + dy * dy) + (dw * dw + dh * dh)) * ct + dc * dc;
  }
  float ts = 0.0f, per = 0.0f;
#pragma unroll
  for (int c = BOXES * NPRED; c < CH; ++c) {
    float t = st[c], p = sp[c];
    ts += t;
    per += t * t - p * p;
  }
  if (ts == 1.0f) loss += per; // exact-equality mask, as in reference
  return loss;
}

// 128 -> 1 block reduction using v_wmma_f32_16x16x4_f32 with B = ones.
// D[m][n] = sum_k A[m][k]; chaining 2 WMMAs over C sums all 128 values into
// every column of D. Per the 16x16 f32 C/D layout, lane L(<16) holds rows
// 0..7 of column L and lane L+16 rows 8..15, so total = s(lane0)+s(lane16).
__device__ __forceinline__ float block_reduce_wmma(float v, int tid) {
  __shared__ float red[BLOCK];
  red[tid] = v;
  __syncthreads();
  float tot = 0.0f;
  if (tid < 32) { // wave 0, EXEC all ones (WMMA requirement)
    v2f ones = {1.0f, 1.0f};
    v8f c = {0.0f, 0.0f, 0.0f, 0.0f, 0.0f, 0.0f, 0.0f, 0.0f};
#pragma unroll
    for (int j = 0; j < BLOCK / 64; ++j) { // 2 chunks of 64 values
      v2f a;
      a[0] = red[j * 64 + tid];
      a[1] = red[j * 64 + 32 + tid];
      c = __builtin_amdgcn_wmma_f32_16x16x4_f32(
          /*neg_a=*/false, a, /*neg_b=*/false, ones,
          /*c_mod=*/(short)0, c, /*reuse_a=*/false, /*reuse_b=*/false);
    }
    float s = 0.0f;
#pragma unroll
    for (int j = 0; j < 8; ++j) s += c[j];
    red[tid] = s;                 // same-wave DS ops are in order
    tot = red[0] + red[16];
  }
  __syncthreads();
  return tot;
}

__global__ __launch_bounds__(BLOCK) void yolo_stage1(
    const float* __restrict__ pred, const float* __restrict__ tgt,
    float* __restrict__ partials) {
  __shared__ __align__(16) float shp[2][TILE_FLOATS]; // 15360 B x2
  __shared__ __align__(16) float sht[2][TILE_FLOATS]; // 15360 B x2 (~62 KB)
  const int tid = threadIdx.x;
  const int tile0 = blockIdx.x * TILES_PER_BLOCK;

  issue_tile(pred, tgt, tile0, shp[0], sht[0], tid);

  float acc = 0.0f;
#pragma unroll
  for (int k = 0; k < TILES_PER_BLOCK; ++k) {
    const int buf = k & 1;
    if (k + 1 < TILES_PER_BLOCK) {
      // prefetch next tile into the other buffer (safe: trailing barrier of
      // the previous iteration guarantees everyone finished with it)
      issue_tile(pred, tgt, tile0 + k + 1, shp[buf ^ 1], sht[buf ^ 1], tid);
      WAIT_ASYNCCNT(ASYNC_PER_TILE); // current tile landed; next in flight
    } else {
      WAIT_ASYNCCNT(0);
    }
    __syncthreads(); // all waves' DMA data visible
    acc += cell_loss(&shp[buf][tid * CH], &sht[buf][tid * CH]);
    __syncthreads(); // done reading before this buffer is refilled
  }

  float tot = block_reduce_wmma(acc, tid);
  if (tid == 0) partials[blockIdx.x] = tot;
}

__global__ __launch_bounds__(BLOCK) void yolo_stage2(
    const float* __restrict__ partials, float* __restrict__ out) {
  const int tid = threadIdx.x;
  float a = 0.0f;
#pragma unroll
  for (int i = 0; i < GRID / BLOCK; ++i) a += partials[i * BLOCK + tid];
  float tot = block_reduce_wmma(a, tid);
  if (tid == 0) out[0] = tot;
}

extern "C" void kernel_launch(void* const* d_in, const int* in_sizes, int n_in,
                              void* d_out, int out_size, void* d_ws,
                              size_t ws_size, hipStream_t stream) {
  (void)in_sizes; (void)n_in; (void)out_size; (void)ws_size;
  const float* pred = (const float*)d_in[0];
  const float* tgt = (const float*)d_in[1];
  float* partials = (float*)d_ws; // GRID floats = 8 KB
  yolo_stage1<<<GRID, BLOCK, 0, stream>>>(pred, tgt, partials);
  yolo_stage2<<<1, BLOCK, 0, stream>>>(partials, (float*)d_out);
}